// QueryAndGroup_27711128994377
// MI455X (gfx1250) — compile-verified
//
#include <hip/hip_runtime.h>
#include <hip/hip_bf16.h>

typedef __attribute__((ext_vector_type(2))) float v2f;
typedef __attribute__((ext_vector_type(8))) float v8f;

#define BB   4
#define NN   8192
#define MM   2048
#define CC   64
#define NS   32
#define R2   0.04f   // radius^2 = 0.2^2

// ---------------------------------------------------------------------------
// Kernel 1: ball query via V_WMMA_F32_16X16X4_F32.
// One wave handles a tile of 16 centers; loops over N in tiles of 16 points.
// D = A x B + C  with
//   A[m] = [-2cx, -2cy, -2cz, 1]        (16x4, 2 VGPRs/lane)
//   B[:,n] = [px, py, pz, |p|^2]        (4x16, 2 VGPRs/lane)
//   C[m][n] = |c_m|^2                   (16x16, 8 VGPRs/lane)
// => D[m][n] = d^2(center m, point n).
// Per D-row ballot gives the in-ball column mask; lane l<16 owns row l and
// appends indices in ascending order (matches stable-argsort reference).
// Early exit once every row has NS hits (wave-uniform vote).
// ---------------------------------------------------------------------------
__global__ void __launch_bounds__(256)
ballquery_wmma_kernel(const float* __restrict__ xyz,
                      const float* __restrict__ new_xyz,
                      int* __restrict__ idx)
{
    const int lane = threadIdx.x & 31;
    const int wave = threadIdx.x >> 5;
    const int tile = blockIdx.x * 8 + wave;     // 512 tiles = B * (M/16)
    const int b    = tile >> 7;                 // tile / 128
    const int m0   = (tile & 127) << 4;         // first center of tile

    const int hi = lane >> 4;                   // 0: lanes 0-15, 1: lanes 16-31
    const int lm = lane & 15;

    const float* cbase = new_xyz + ((size_t)b * MM + m0) * 3;

    // A matrix: row m = lm; lanes 0-15 carry K=0,1 ; lanes 16-31 carry K=2,3
    float cx = cbase[lm * 3 + 0];
    float cy = cbase[lm * 3 + 1];
    float cz = cbase[lm * 3 + 2];
    v2f a;
    a.x = hi ? (-2.0f * cz) : (-2.0f * cx);
    a.y = hi ? 1.0f         : (-2.0f * cy);

    // C matrix: VGPR k holds row m = k + 8*hi -> |center|^2 (col-independent)
    v8f cacc;
#pragma unroll
    for (int k = 0; k < 8; ++k) {
        int m = k + 8 * hi;
        float x = cbase[m * 3 + 0];
        float y = cbase[m * 3 + 1];
        float z = cbase[m * 3 + 2];
        cacc[k] = x * x + y * y + z * z;
    }

    int cnt = 0, first = 0;
    const float* pbase = xyz + (size_t)b * NN * 3;
    int* rowout = idx + ((size_t)(b * MM + m0 + lm)) * NS;   // row owned by lane<16

    for (int nt = 0; nt < NN / 16; ++nt) {
        // Wave-uniform early exit: stop once all 16 rows have NS hits.
        // (lanes 16-31 never count -> contribute 0 to the vote)
        if (__builtin_amdgcn_ballot_w32((lane < 16) && (cnt < NS)) == 0u)
            break;

        // B matrix: col n = lm; lanes 0-15 carry K=0,1 ; lanes 16-31 carry K=2,3
        // Branchless build: no exec-mask save/restore in the hot loop.
        int n = nt * 16 + lm;
        float px  = pbase[n * 3 + 0];
        float py  = pbase[n * 3 + 1];
        float pz  = pbase[n * 3 + 2];
        float nsq = px * px + py * py + pz * pz;
        v2f bv;
        bv.x = hi ? pz  : px;
        bv.y = hi ? nsq : py;

        // Full-EXEC here (all 256 threads live) as WMMA requires.
        v8f d = __builtin_amdgcn_wmma_f32_16x16x4_f32(
                    /*neg_a=*/false, a, /*neg_b=*/false, bv,
                    /*c_mod=*/(short)0, cacc,
                    /*reuse_a=*/false, /*reuse_b=*/false);

        // Per-row in-ball mask: ballot over d[k] covers rows k (bits 0-15)
        // and k+8 (bits 16-31). Route each row's 16 bits to its owner lane.
        unsigned bits = 0;
#pragma unroll
        for (int k = 0; k < 8; ++k) {
            unsigned bk = __builtin_amdgcn_ballot_w32(d[k] < R2);
            if ((lane & 7) == k)
                bits = (lane < 8) ? (bk & 0xFFFFu) : (bk >> 16);
        }

        if (lane < 16) {
            while (bits) {
                int col  = __builtin_ctz(bits);
                bits    &= bits - 1;
                int pidx = nt * 16 + col;
                if (cnt == 0) first = pidx;
                if (cnt < NS) rowout[cnt] = pidx;
                ++cnt;
            }
        }
    }

    if (lane < 16) {
        for (int j = cnt; j < NS; ++j) rowout[j] = first;   // pad with first (0 if empty)
    }
}

// ---------------------------------------------------------------------------
// Kernel 2: gather + concat. One block per (b, m). threadIdx.x = sample slot
// (32 consecutive output floats -> 128B coalesced stores per channel),
// threadIdx.y strides over the 67 output channels.
// out layout: (B, 3+C, M, NS)
// ---------------------------------------------------------------------------
__global__ void __launch_bounds__(256)
group_gather_kernel(const float* __restrict__ xyz,
                    const float* __restrict__ new_xyz,
                    const float* __restrict__ feat,
                    const int* __restrict__ idx,
                    float* __restrict__ out)
{
    const int bm = blockIdx.x;          // 0 .. B*M-1
    const int b  = bm >> 11;            // / 2048
    const int m  = bm & (MM - 1);
    const int j  = threadIdx.x;         // 0..31 sample slot

    const int id = idx[(size_t)bm * NS + j];

    for (int c = threadIdx.y; c < 3 + CC; c += 8) {
        float v;
        if (c < 3) {
            v = xyz[((size_t)b * NN + id) * 3 + c]
              - new_xyz[((size_t)b * MM + m) * 3 + c];
        } else {
            v = feat[((size_t)b * CC + (c - 3)) * NN + id];
        }
        out[(((size_t)b * (3 + CC) + c) * MM + m) * NS + j] = v;
    }
}

extern "C" void kernel_launch(void* const* d_in, const int* in_sizes, int n_in,
                              void* d_out, int out_size, void* d_ws, size_t ws_size,
                              hipStream_t stream)
{
    const float* xyz      = (const float*)d_in[0];  // (B, N, 3)
    const float* new_xyz  = (const float*)d_in[1];  // (B, M, 3)
    const float* features = (const float*)d_in[2];  // (B, C, N)
    float*       out      = (float*)d_out;          // (B, 3+C, M, NS)

    int* idx = (int*)d_ws;                          // (B, M, NS) = 1 MB scratch

    // Kernel 1: 512 center tiles, 8 waves (256 thr) per block -> 64 blocks.
    ballquery_wmma_kernel<<<(BB * (MM / 16)) / 8, 256, 0, stream>>>(
        xyz, new_xyz, idx);

    // Kernel 2: one block per (b, m).
    dim3 blk(32, 8);
    group_gather_kernel<<<BB * MM, blk, 0, stream>>>(
        xyz, new_xyz, features, idx, out);

    (void)in_sizes; (void)n_in; (void)out_size; (void)ws_size;
}